// FactoredMPS_25881472925903
// MI455X (gfx1250) — compile-verified
//
#include <hip/hip_runtime.h>
#include <math.h>

#define LDS_FENCE() asm volatile("s_wait_dscnt 0" ::: "memory")

typedef __attribute__((ext_vector_type(2))) float v2f;
typedef __attribute__((ext_vector_type(8))) float v8f;

static constexpr int kNumCores = 64;
static constexpr int kVocab    = 8192;

// height-3 boundary core contraction (h_dim = 4): rows(4x16 per elem) -> out(64 per elem)
__device__ __forceinline__ void v_contract_h4(const float* rows, const float* k1b,
                                              const float* topb, float* outw, int lane) {
  const int e = lane >> 3;
  const int base = (lane & 7) * 8;
#pragma unroll
  for (int t = 0; t < 8; ++t) {
    const int j = base + t;
    const int p = j >> 2, wq = j & 3;
    float acc = 0.f;
#pragma unroll
    for (int q = 0; q < 4; ++q) {
      const int f = 4 * p + q;
      const int m = f >> 4, n = f & 15;
      float tmp = 0.f;
#pragma unroll
      for (int vb = 0; vb < 4; ++vb)
        tmp += rows[e * 16 + 4 * m + vb] * k1b[vb * 16 + n];
      acc += tmp * topb[q * 4 + wq];
    }
    outw[e * 64 + j] = acc;
  }
}

__global__ __launch_bounds__(256)
void mps_chain_kernel(const int* __restrict__ x,
                      const float* __restrict__ bnd_emb,
                      const float* __restrict__ bnd_k1,
                      const float* __restrict__ bnd_top,
                      const float* __restrict__ mid_emb,
                      const float* __restrict__ mid_k1,
                      const float* __restrict__ mid_top,
                      float* __restrict__ out, int Btot) {
  __shared__ float sL[8][256];    // L per wave: 4 elems x 64
  __shared__ float sEmb[8][256];  // gathered emb rows / right-boundary R
  __shared__ float sK1[8][256];   // 4x64 k1 (or 4x16 boundary k1)
  __shared__ float sTop[8][64];   // 4x16 top (or 4x4 boundary top)
  __shared__ float sG[8][64];     // G[e][d*4+vb] (also boundary rows buffer)
  __shared__ float sKk[8][64];    // Kk[e][c1*4+q]
  __shared__ float sLs[8][4];     // log-scalar accumulators
  __shared__ int   sTok[8][256];  // 4 elems x 64 tokens

  const int lane = threadIdx.x & 31;
  const int wv   = threadIdx.x >> 5;
  const int be0  = (blockIdx.x * 8 + wv) * 4;
  if (be0 + 4 > Btot) return;

  float* LW = sL[wv];   float* embW = sEmb[wv]; float* k1W = sK1[wv];
  float* topW = sTop[wv]; float* gW = sG[wv];   float* kkW = sKk[wv];
  float* lsW = sLs[wv]; int* tokW = sTok[wv];

  // stage all 4 elements' tokens (256 contiguous ints)
  {
    const int4* src = (const int4*)(x + (size_t)be0 * kNumCores);
    ((int4*)tokW)[2 * lane]     = src[2 * lane];
    ((int4*)tokW)[2 * lane + 1] = src[2 * lane + 1];
  }
  if (lane < 4) lsW[lane] = 0.f;
  LDS_FENCE();

  const int hi  = lane >> 4;  // lane half (WMMA register layouts)
  const int col = lane & 15;

  // ---- left boundary core -> L0 ----
  {
    if (lane >= 16) {
      const int l2 = lane - 16;
      const int e = l2 >> 2, off = (l2 & 3) * 4;
      const int tok = tokW[e * kNumCores + 0];
      *(float4*)&gW[e * 16 + off] =
          *(const float4*)(bnd_emb + ((size_t)tok) * 16 + off);
    } else {
      *(float4*)&k1W[4 * lane] = *(const float4*)(bnd_k1 + 4 * lane);
      if (lane < 4)
        *(float4*)&topW[4 * lane] = *(const float4*)(bnd_top + 4 * lane);
    }
    LDS_FENCE();
    v_contract_h4(gW, k1W, topW, LW, lane);
    LDS_FENCE();
  }

  // ---- main chain: i = 1 .. 62 ----
  for (int i = 1; i <= kNumCores - 2; ++i) {
    const int s = i - 1;
    {  // stage k1 (4x64)
      const float4* ksrc = (const float4*)(mid_k1 + (size_t)s * 256);
      ((float4*)k1W)[2 * lane]     = ksrc[2 * lane];
      ((float4*)k1W)[2 * lane + 1] = ksrc[2 * lane + 1];
    }
    if (lane < 16)  // stage top (4x16)
      ((float4*)topW)[lane] = ((const float4*)(mid_top + (size_t)s * 64))[lane];
    {  // gather emb row per element (4 x 64 floats)
      const int e = lane >> 3, off = (lane & 7) * 8;
      const int tok = tokW[e * kNumCores + i];
      const float* src = mid_emb + ((size_t)s * kVocab + tok) * 64 + off;
      *(float4*)&embW[e * 64 + off]     = *(const float4*)src;
      *(float4*)&embW[e * 64 + off + 4] = *(const float4*)(src + 4);
    }
    if (i < kNumCores - 2 && (lane & 7) == 0) {  // prefetch next gather
      const int e = lane >> 3;
      const int ntok = tokW[e * kNumCores + i + 1];
      __builtin_prefetch(mid_emb + ((size_t)i * kVocab + ntok) * 64, 0, 0);
    }
    LDS_FENCE();

    // step A (VALU): G_e[d,vb] = sum_a L_e[4a+d] * emb_e[4a+vb]
#pragma unroll
    for (int t = 0; t < 2; ++t) {
      const int g = 2 * lane + t;
      const int e = g >> 4, rem = g & 15, dd = rem >> 2, vb = rem & 3;
      const float* Le = &LW[e * 64];
      const float* ee = &embW[e * 64];
      float acc = 0.f;
#pragma unroll
      for (int a = 0; a < 16; ++a) acc += Le[4 * a + dd] * ee[4 * a + vb];
      gW[g] = acc;
    }
    LDS_FENCE();

    // WMMA stage 1: O_j = G-stack(16x4) x k1[:,16j:16j+16]; Kk accumulated in regs
    v2f A;
    A.x = gW[4 * col + 2 * hi];      // A[r=col, vb=2*hi]
    A.y = gW[4 * col + 2 * hi + 1];  // vb=2*hi+1
    float acc0 = 0.f, acc1 = 0.f;
#pragma unroll
    for (int j = 0; j < 4; ++j) {
      v2f Bm;
      Bm.x = k1W[(2 * hi) * 64 + 16 * j + col];
      Bm.y = k1W[(2 * hi + 1) * 64 + 16 * j + col];
      v8f C = {};
      C = __builtin_amdgcn_wmma_f32_16x16x4_f32(false, A, false, Bm,
                                                (short)0, C, false, false);
      acc0 += C[j];      // row 8*hi+j   -> element 2*hi
      acc1 += C[j + 4];  // row 8*hi+4+j -> element 2*hi+1
    }
    kkW[(2 * hi) * 16 + col]     = acc0;
    kkW[(2 * hi + 1) * 16 + col] = acc1;
    LDS_FENCE();

    // WMMA stage 2: L' = Kk-stack(16x4) x top(4x16)
    v2f A2, B2;
    A2.x = kkW[4 * col + 2 * hi];
    A2.y = kkW[4 * col + 2 * hi + 1];
    B2.x = topW[(2 * hi) * 16 + col];
    B2.y = topW[(2 * hi + 1) * 16 + col];
    v8f D = {};
    D = __builtin_amdgcn_wmma_f32_16x16x4_f32(false, A2, false, B2,
                                              (short)0, D, false, false);
#pragma unroll
    for (int v = 0; v < 8; ++v) {
      const int r = v + 8 * hi;  // r = 4e + c1
      LW[(r >> 2) * 64 + (r & 3) * 16 + col] = D[v];
    }
    LDS_FENCE();

    if ((i & 7) == 0) {  // rescale every PARTIAL=8 steps
      const int e = lane >> 3, k0 = (lane & 7) * 8;
      float ss = 0.f;
#pragma unroll
      for (int t = 0; t < 8; ++t) { const float v = LW[e * 64 + k0 + t]; ss += v * v; }
      ss += __shfl_xor(ss, 1, 32);
      ss += __shfl_xor(ss, 2, 32);
      ss += __shfl_xor(ss, 4, 32);
      const float nrm = sqrtf(ss);
      const float inv = 1.f / nrm;
#pragma unroll
      for (int t = 0; t < 8; ++t) LW[e * 64 + k0 + t] *= inv;
      if ((lane & 7) == 0) lsW[e] += logf(nrm);
      LDS_FENCE();
    }
  }

  // ---- right boundary core -> R, dot with L, emit log-scalar ----
  {
    if (lane >= 16) {
      const int l2 = lane - 16;
      const int e = l2 >> 2, off = (l2 & 3) * 4;
      const int tok = tokW[e * kNumCores + (kNumCores - 1)];
      *(float4*)&gW[e * 16 + off] =
          *(const float4*)(bnd_emb + ((size_t)kVocab + tok) * 16 + off);
    } else {
      *(float4*)&k1W[4 * lane] = *(const float4*)(bnd_k1 + 64 + 4 * lane);
      if (lane < 4)
        *(float4*)&topW[4 * lane] = *(const float4*)(bnd_top + 16 + 4 * lane);
    }
    LDS_FENCE();
    v_contract_h4(gW, k1W, topW, embW, lane);
    LDS_FENCE();

    const int e = lane >> 3, k0 = (lane & 7) * 8;
    float p = 0.f;
#pragma unroll
    for (int t = 0; t < 8; ++t) p += LW[e * 64 + k0 + t] * embW[e * 64 + k0 + t];
    p += __shfl_xor(p, 1, 32);
    p += __shfl_xor(p, 2, 32);
    p += __shfl_xor(p, 4, 32);
    if ((lane & 7) == 0) out[be0 + e] = lsW[e] + logf(fabsf(p));
  }
}

extern "C" void kernel_launch(void* const* d_in, const int* in_sizes, int n_in,
                              void* d_out, int out_size, void* d_ws, size_t ws_size,
                              hipStream_t stream) {
  (void)n_in; (void)d_ws; (void)ws_size; (void)out_size;
  const int*   xp      = (const int*)d_in[0];
  const float* bnd_emb = (const float*)d_in[1];
  const float* bnd_k1  = (const float*)d_in[2];
  const float* bnd_top = (const float*)d_in[3];
  const float* mid_emb = (const float*)d_in[4];
  const float* mid_k1  = (const float*)d_in[5];
  const float* mid_top = (const float*)d_in[6];
  float* out = (float*)d_out;
  const int Bt = in_sizes[0] / kNumCores;  // 1024
  const int elemsPerBlock = 32;            // 8 waves x 4 elements
  dim3 grid((Bt + elemsPerBlock - 1) / elemsPerBlock), block(256);
  hipLaunchKernelGGL(mps_chain_kernel, grid, block, 0, stream,
                     xp, bnd_emb, bnd_k1, bnd_top, mid_emb, mid_k1, mid_top, out, Bt);
}